// TemporalAttention_60318520705653
// MI455X (gfx1250) — compile-verified
//
#include <hip/hip_runtime.h>
#include <hip/hip_bf16.h>
#include <math.h>

typedef __attribute__((ext_vector_type(2))) float v2f;
typedef __attribute__((ext_vector_type(8))) float v8f;

#define S_LEN   2048
#define BATCH   4
#define DMODEL  512
#define NHEAD   8
#define HDIM    64
#define MROWS   (BATCH * S_LEN)   /* 8192 */
#define MAXLAG  10
#define WIN     11

/* ------------------------------------------------------------------ */
/*  GEMM: C[M,512] = A[M,512] @ W[512,512] + bias   (fp32 WMMA 16x16x4) */
/*  Double-buffered LDS; A tile staged via GLOBAL_LOAD_ASYNC_TO_LDS.    */
/* ------------------------------------------------------------------ */
#define BM 128
#define BN 64
#define BK 16
#define LDSS 18   /* padded LDS row stride: even -> 8B-aligned b64, gcd(18,64)=2 -> conflict-free 16-row reads */
#define KTILES (DMODEL / BK)      /* 32 */

__global__ __launch_bounds__(256) void gemm_wmma_f32(
    const float* __restrict__ A, const float* __restrict__ W,
    const float* __restrict__ bias, float* __restrict__ C)
{
    __shared__ float As[2][BM * LDSS];   /* 2 x 9216 B */
    __shared__ float Bs[2][BN * LDSS];   /* 2 x 4608 B */

    const int tid   = threadIdx.x;
    const int wave  = tid >> 5;
    const int lane  = tid & 31;
    const int half  = lane >> 4;     /* 0: K 0..1 / rows M..M+7 ; 1: K 2..3 / rows M+8.. */
    const int lr    = lane & 15;
    const int waveM = wave >> 1;     /* 0..3 */
    const int waveN = wave & 1;      /* 0..1 */

    const int rowBase = blockIdx.y * BM;
    const int colBase = blockIdx.x * BN;

    /* per-thread A-tile async-copy coordinates: 4 ops x 2 floats = 8 floats/thread */
    int arow[4], akp[4];
#pragma unroll
    for (int o = 0; o < 4; ++o) {
        const int e = tid + o * 256;        /* 0..1023 float2 slots of the 128x16 tile */
        arow[o] = e >> 3;
        akp[o]  = (e & 7) << 1;
    }
    /* per-thread W-tile coordinates (16x64, transposed into Bs[n][k]) */
    const int wk = tid >> 4;                /* 0..15 */
    const int wn = (tid & 15) << 2;         /* 0..60 */

    const v8f vzero = {0.f,0.f,0.f,0.f,0.f,0.f,0.f,0.f};
    v8f acc[2][2];
    acc[0][0] = vzero; acc[0][1] = vzero; acc[1][0] = vzero; acc[1][1] = vzero;

#define STAGE_A_ASYNC(P, K0)                                                     \
    {                                                                            \
        _Pragma("unroll")                                                        \
        for (int o = 0; o < 4; ++o) {                                            \
            const float* src = A + (size_t)(rowBase + arow[o]) * DMODEL          \
                                 + (K0) + akp[o];                                \
            unsigned dst = (unsigned)(uintptr_t)&As[P][arow[o] * LDSS + akp[o]]; \
            asm volatile("global_load_async_to_lds_b64 %0, %1, off"              \
                         :: "v"(dst), "v"(src) : "memory");                      \
        }                                                                        \
    }

#define STAGE_W(P, K0)                                                           \
    {                                                                            \
        const float4 f = *(const float4*)(W + (size_t)((K0) + wk) * DMODEL       \
                                            + colBase + wn);                     \
        Bs[P][(wn + 0) * LDSS + wk] = f.x;                                       \
        Bs[P][(wn + 1) * LDSS + wk] = f.y;                                       \
        Bs[P][(wn + 2) * LDSS + wk] = f.z;                                       \
        Bs[P][(wn + 3) * LDSS + wk] = f.w;                                       \
    }

    /* prologue: fill buffer 0 */
    STAGE_W(0, 0)
    STAGE_A_ASYNC(0, 0)
    asm volatile("s_wait_asynccnt 0x0" ::: "memory");
    __syncthreads();

    for (int kt = 0; kt < KTILES; ++kt) {
        const int p = kt & 1;
        if (kt + 1 < KTILES) {
            STAGE_W(p ^ 1, (kt + 1) * BK)
            STAGE_A_ASYNC(p ^ 1, (kt + 1) * BK)
        }

#pragma unroll
        for (int ks = 0; ks < 4; ++ks) {
            const int kk = ks * 4 + half * 2;
            v2f a0 = *(const v2f*)&As[p][(waveM * 32 +  0 + lr) * LDSS + kk];
            v2f a1 = *(const v2f*)&As[p][(waveM * 32 + 16 + lr) * LDSS + kk];
            v2f b0 = *(const v2f*)&Bs[p][(waveN * 32 +  0 + lr) * LDSS + kk];
            v2f b1 = *(const v2f*)&Bs[p][(waveN * 32 + 16 + lr) * LDSS + kk];
            acc[0][0] = __builtin_amdgcn_wmma_f32_16x16x4_f32(false, a0, false, b0, (short)0, acc[0][0], false, false);
            acc[0][1] = __builtin_amdgcn_wmma_f32_16x16x4_f32(false, a0, false, b1, (short)0, acc[0][1], false, false);
            acc[1][0] = __builtin_amdgcn_wmma_f32_16x16x4_f32(false, a1, false, b0, (short)0, acc[1][0], false, false);
            acc[1][1] = __builtin_amdgcn_wmma_f32_16x16x4_f32(false, a1, false, b1, (short)0, acc[1][1], false, false);
        }

        asm volatile("s_wait_asynccnt 0x0" ::: "memory");
        __syncthreads();
    }

    /* epilogue: C/D layout: VGPR v, lanes 0-15 -> (M=v, N=lr); lanes 16-31 -> (M=v+8, N=lr) */
#pragma unroll
    for (int mt = 0; mt < 2; ++mt)
#pragma unroll
        for (int nt = 0; nt < 2; ++nt) {
            const int col = colBase + waveN * 32 + nt * 16 + lr;
            const float bv = bias[col];
#pragma unroll
            for (int vi = 0; vi < 8; ++vi) {
                const int row = rowBase + waveM * 32 + mt * 16 + vi + half * 8;
                C[(size_t)row * DMODEL + col] = acc[mt][nt][vi] + bv;
            }
        }
#undef STAGE_A_ASYNC
#undef STAGE_W
}

/* ------------------------------------------------------------------ */
/*  Banded attention: one wave per (b,i,h); window of <=11 keys        */
/* ------------------------------------------------------------------ */
__global__ __launch_bounds__(256) void band_attn(
    const float* __restrict__ q, const float* __restrict__ k,
    const float* __restrict__ v, const float* __restrict__ decp,
    float* __restrict__ ctx, float* __restrict__ attnw)
{
    const int tid  = threadIdx.x;
    const int wave = tid >> 5;
    const int lane = tid & 31;
    const int gw   = blockIdx.x * 8 + wave;      /* 0 .. B*S*H-1 = 65535 */
    const int h = gw & 7;
    const int i = (gw >> 3) & (S_LEN - 1);
    const int b = gw >> 14;

    const size_t rowOff = (size_t)(b * S_LEN + i) * DMODEL + h * HDIM + lane * 2;
    const float qx = q[rowOff + 0];
    const float qy = q[rowOff + 1];

    /* temporal decay: d[lag] = -log(softplus(p[lag]) + 1e-8) */
    float dval[WIN];
#pragma unroll
    for (int l = 0; l < WIN; ++l) {
        float p  = decp[l];
        float sp = (p > 20.f) ? p : log1pf(expf(p));
        dval[l]  = -logf(sp + 1e-8f);
    }

    const int jstart = (i >= MAXLAG) ? (i - MAXLAG) : 0;
    const int nj     = i - jstart + 1;

    float s[WIN];
#pragma unroll
    for (int idx = 0; idx < WIN; ++idx) {
        if (idx < nj) {
            const int j = jstart + idx;
            const size_t ko = (size_t)(b * S_LEN + j) * DMODEL + h * HDIM + lane * 2;
            float part = qx * k[ko] + qy * k[ko + 1];
            part += __shfl_xor(part, 16);
            part += __shfl_xor(part, 8);
            part += __shfl_xor(part, 4);
            part += __shfl_xor(part, 2);
            part += __shfl_xor(part, 1);
            s[idx] = part * 0.125f + dval[i - j];
        } else {
            s[idx] = -1e30f;
        }
    }

    float m = s[0];
#pragma unroll
    for (int idx = 1; idx < WIN; ++idx) m = fmaxf(m, s[idx]);
    float sum = 0.f, w[WIN];
#pragma unroll
    for (int idx = 0; idx < WIN; ++idx) { w[idx] = expf(s[idx] - m); sum += w[idx]; }
    const float inv = 1.f / sum;
#pragma unroll
    for (int idx = 0; idx < WIN; ++idx) w[idx] *= inv;

    /* ctx = sum_j w[j] * v[j] */
    float cx = 0.f, cy = 0.f;
#pragma unroll
    for (int idx = 0; idx < WIN; ++idx) {
        if (idx < nj) {
            const int j = jstart + idx;
            const size_t vo = (size_t)(b * S_LEN + j) * DMODEL + h * HDIM + lane * 2;
            cx += w[idx] * v[vo];
            cy += w[idx] * v[vo + 1];
        }
    }
    ctx[rowOff + 0] = cx;
    ctx[rowOff + 1] = cy;

    /* per-head window weights (indexed by window position), lanes 0..10 store */
    const size_t ab = (size_t)((b * NHEAD + h) * S_LEN + i) * WIN;
#pragma unroll
    for (int idx = 0; idx < WIN; ++idx) {
        if (lane == idx) attnw[ab + idx] = (idx < nj) ? w[idx] : 0.f;
    }
}

/* ------------------------------------------------------------------ */
/*  attn.mean(axis=1) -> dense [B,S,S] (zeros outside the 11-band)     */
/* ------------------------------------------------------------------ */
__global__ __launch_bounds__(256) void attn_mean_fill(
    const float* __restrict__ attnw, float* __restrict__ out)
{
    const unsigned idx = blockIdx.x * 256u + threadIdx.x;   /* < 2^24 */
    const int j = idx & (S_LEN - 1);
    const int i = (idx >> 11) & (S_LEN - 1);
    const int b = idx >> 22;
    const int lag = i - j;
    float val = 0.f;
    if (lag >= 0 && lag <= MAXLAG) {
        const int jstart = (i >= MAXLAG) ? (i - MAXLAG) : 0;
        const int wi = j - jstart;
        float sum = 0.f;
#pragma unroll
        for (int h = 0; h < NHEAD; ++h)
            sum += attnw[(size_t)((b * NHEAD + h) * S_LEN + i) * WIN + wi];
        val = sum * (1.f / NHEAD);
    }
    out[idx] = val;
}

/* ------------------------------------------------------------------ */
extern "C" void kernel_launch(void* const* d_in, const int* in_sizes, int n_in,
                              void* d_out, int out_size, void* d_ws, size_t ws_size,
                              hipStream_t stream)
{
    (void)in_sizes; (void)n_in; (void)out_size; (void)ws_size;
    const float* x   = (const float*)d_in[0];
    const float* Wq  = (const float*)d_in[1];
    const float* bq  = (const float*)d_in[2];
    const float* Wk  = (const float*)d_in[3];
    const float* bk  = (const float*)d_in[4];
    const float* Wv  = (const float*)d_in[5];
    const float* bv  = (const float*)d_in[6];
    const float* Wo  = (const float*)d_in[7];
    const float* bo  = (const float*)d_in[8];
    const float* dec = (const float*)d_in[9];

    float* out      = (float*)d_out;                         /* [B,S,D] */
    float* out_attn = out + (size_t)MROWS * DMODEL;          /* [B,S,S] */

    float* qws = (float*)d_ws;
    float* kws = qws + (size_t)MROWS * DMODEL;
    float* vws = kws + (size_t)MROWS * DMODEL;
    float* cws = vws + (size_t)MROWS * DMODEL;
    float* aw  = cws + (size_t)MROWS * DMODEL;               /* [B,H,S,11] */

    dim3 gg(DMODEL / BN, MROWS / BM);                        /* 8 x 64 */
    gemm_wmma_f32<<<gg, 256, 0, stream>>>(x,   Wq, bq, qws);
    gemm_wmma_f32<<<gg, 256, 0, stream>>>(x,   Wk, bk, kws);
    gemm_wmma_f32<<<gg, 256, 0, stream>>>(x,   Wv, bv, vws);

    band_attn<<<(BATCH * S_LEN * NHEAD) / 8, 256, 0, stream>>>(qws, kws, vws, dec, cws, aw);

    gemm_wmma_f32<<<gg, 256, 0, stream>>>(cws, Wo, bo, out);

    attn_mean_fill<<<(BATCH * S_LEN * S_LEN) / 256, 256, 0, stream>>>(aw, out_attn);
}